// Embedding_30863634989540
// MI455X (gfx1250) — compile-verified
//
#include <hip/hip_runtime.h>
#include <hip/hip_bf16.h>

// Embedding gather: out[b,w,:] = (ids[b,w] != 0) ? table[ids[b,w],:] : 0
// BATCH=4096, WIDTH=200, DIM=64, VOCAB=1e6. fp32 table/output, int ids.
//
// Memory-bound op: ~420 MB total traffic -> ~18us floor at 23.3 TB/s.
// Strategy: 16 lanes per row, one float4 (16B) per lane -> each half-wave
// moves one contiguous 256B row with global_load_b128/global_store_b128.
// Output stores are non-temporal (TH=NT) so the 200MiB streaming output
// does not evict hot table rows from the 192MB L2.

typedef float v4f __attribute__((ext_vector_type(4)));

#define DIM 64
#define VEC 4
#define LANES_PER_ROW (DIM / VEC)   // 16

__global__ __launch_bounds__(256) void
embedding_gather_kernel(const int* __restrict__ ids,
                        const float* __restrict__ table,
                        float* __restrict__ out,
                        int rows) {
    const long tid = (long)blockIdx.x * blockDim.x + threadIdx.x;
    const long row   = tid >> 4;          // which (b,w) token
    const int  chunk = (int)(tid & 15);   // which 16B slice of the 256B row

    if (row >= rows) return;

    // 16 lanes read the same 4B id word; the coalescer merges this into a
    // single request. Regular-temporal: ids are tiny (3.3MB) and reused.
    const int id = ids[row];

    // Branch-free padding mask: always issue the b128 load (id==0 reads
    // row 0, which is valid memory), then select zero. Keeps the memory
    // pipeline uniform across the wave — no divergence on the sentinel.
    const v4f* __restrict__ src =
        (const v4f*)table + ((long)id * LANES_PER_ROW + chunk);
    v4f v = *src;   // global_load_b128, temporal (keep table rows in L2)

    if (id == 0) {
        v = (v4f){0.0f, 0.0f, 0.0f, 0.0f};
    }

    // Streaming write-once output: non-temporal b128 store (TH=NT) so the
    // 200MiB output stream doesn't thrash the L2 working set of the table.
    v4f* __restrict__ dst = (v4f*)out + (row * LANES_PER_ROW + chunk);
    __builtin_nontemporal_store(v, dst);
}

extern "C" void kernel_launch(void* const* d_in, const int* in_sizes, int n_in,
                              void* d_out, int out_size, void* d_ws, size_t ws_size,
                              hipStream_t stream) {
    const int*   ids   = (const int*)d_in[0];   // input_ids, flat [B*W]
    const float* table = (const float*)d_in[1]; // [VOCAB, 64]
    float*       out   = (float*)d_out;         // [B*W, 64]

    const int rows = in_sizes[0];               // 4096*200 = 819200

    const long total_threads = (long)rows * LANES_PER_ROW;  // 13,107,200
    const int  block = 256;                                  // 8 wave32s
    const int  grid  = (int)((total_threads + block - 1) / block);

    embedding_gather_kernel<<<grid, block, 0, stream>>>(ids, table, out, rows);
}